// GraphConvLayer_17162689314845
// MI455X (gfx1250) — compile-verified
//
#include <hip/hip_runtime.h>
#include <hip/hip_bf16.h>

typedef __attribute__((ext_vector_type(2))) float v2f;
typedef __attribute__((ext_vector_type(8))) float v8f;
typedef __attribute__((ext_vector_type(4))) float f4;

#define N_NODES 100000
#define FEAT 64
#define NEDGE 1280000

// ---------------------------------------------------------------------------
// Kernel 0: zero the aggregation buffer h (must re-run every call: graph replay)
// ---------------------------------------------------------------------------
__global__ void gc_zero_kernel(float* __restrict__ h, long long n4) {
    long long i = (long long)blockIdx.x * blockDim.x + threadIdx.x;
    if (i < n4) {
        f4 z = {0.f, 0.f, 0.f, 0.f};
        ((f4*)h)[i] = z;
    }
}

// ---------------------------------------------------------------------------
// Kernel 1: edge-parallel scatter-add.  h[dst[e]][f] += x[src[e]][f]
// Each wave32 covers one half-row of one edge (wave-uniform e -> scalar
// src/dst loads), gather of x and atomics to h fully coalesced.  h (25.6 MB)
// is resident in the 192 MB L2, so the 82M f32 atomics resolve on-chip.
// ---------------------------------------------------------------------------
__global__ void gc_scatter_kernel(const float* __restrict__ x,
                                  const int* __restrict__ src,
                                  const int* __restrict__ dst,
                                  float* __restrict__ h) {
    long long idx = (long long)blockIdx.x * blockDim.x + threadIdx.x;
    const long long total = (long long)NEDGE * FEAT;
    if (idx >= total) return;
    int e = (int)(idx >> 6);       // edge index
    int f = (int)(idx & 63);       // feature index
    int s = src[e];
    int d = dst[e];
    float v = x[(long long)s * FEAT + f];
    atomicAdd(&h[(long long)d * FEAT + f], v);
}

// ---------------------------------------------------------------------------
// Kernel 2: fused dual GEMM with full-precision f32 WMMA.
//   out = h @ W_lin^T + x @ W_self^T + (b_lin + b_self + bias)
// One wave owns a 16-row M-strip and computes ALL 4 N-tiles of the 64-wide
// output with 4 accumulators: each A-matrix float2 load (rows of h / x) is
// reused by 4 v_wmma_f32_16x16x4_f32 ops, cutting h/x read traffic 4x vs the
// one-tile-per-wave tiling.  B (16 KB weights) rereads hit L1/L2.
// 128 WMMAs per wave, K=64 per GEMM, exact f32 semantics.
// ---------------------------------------------------------------------------
__global__ __launch_bounds__(256) void gc_gemm_kernel(
    const float* __restrict__ h,      // [N_NODES, 64] aggregated
    const float* __restrict__ x,      // [N_NODES, 64]
    const float* __restrict__ W_lin,  // [64, 64] row-major (o, f)
    const float* __restrict__ W_self, // [64, 64]
    const float* __restrict__ b_lin,  // [64]
    const float* __restrict__ b_self, // [64]
    const float* __restrict__ bias,   // [64]
    float* __restrict__ out)          // [N_NODES, 64]
{
    const int lane = threadIdx.x & 31;
    const int wave = threadIdx.x >> 5;                  // 0..7
    const int m0 = (blockIdx.x * 8 + wave) * 16;        // 16-row M-strip
    if (m0 >= N_NODES) return;                          // wave-uniform: EXEC stays all-1s

    const int r15  = lane & 15;        // A row in tile; also B col (= weight row)
    const int koff = (lane >> 4) << 1; // lanes 0-15 -> K+0..1, lanes 16-31 -> K+2..3

    // Per-lane base pointers (contiguous float2 loads match ISA A/B layouts)
    const float* aLin  = h      + (long long)(m0 + r15) * FEAT + koff;
    const float* aSelf = x      + (long long)(m0 + r15) * FEAT + koff;
    const float* bLin  = W_lin  + (long long)r15 * FEAT + koff;  // + j*16*64 per N-tile
    const float* bSelf = W_self + (long long)r15 * FEAT + koff;

    v8f acc[4] = {v8f{}, v8f{}, v8f{}, v8f{}};

#pragma unroll
    for (int k = 0; k < FEAT; k += 4) {
        v2f a = *(const v2f*)(aLin + k);               // loaded once, used 4x
#pragma unroll
        for (int j = 0; j < 4; ++j) {
            v2f b = *(const v2f*)(bLin + j * (16 * FEAT) + k);
            acc[j] = __builtin_amdgcn_wmma_f32_16x16x4_f32(
                /*neg_a=*/false, a, /*neg_b=*/false, b,
                /*c_mod=*/(short)0, acc[j], /*reuse_a=*/false, /*reuse_b=*/false);
        }
    }
#pragma unroll
    for (int k = 0; k < FEAT; k += 4) {
        v2f a = *(const v2f*)(aSelf + k);
#pragma unroll
        for (int j = 0; j < 4; ++j) {
            v2f b = *(const v2f*)(bSelf + j * (16 * FEAT) + k);
            acc[j] = __builtin_amdgcn_wmma_f32_16x16x4_f32(
                false, a, false, b, (short)0, acc[j], false, false);
        }
    }

    // D layout: VGPR r -> row m0 + r + 8*(lane>=16), col n0 + (lane&15)
    const int mbase = m0 + ((lane >> 4) << 3);
#pragma unroll
    for (int j = 0; j < 4; ++j) {
        const int n = j * 16 + r15;
        const float bsum = b_lin[n] + b_self[n] + bias[n];
#pragma unroll
        for (int r = 0; r < 8; ++r) {
            out[(long long)(mbase + r) * FEAT + n] = acc[j][r] + bsum;
        }
    }
}

// ---------------------------------------------------------------------------
extern "C" void kernel_launch(void* const* d_in, const int* in_sizes, int n_in,
                              void* d_out, int out_size, void* d_ws, size_t ws_size,
                              hipStream_t stream) {
    const float* x      = (const float*)d_in[0];
    const int*   src    = (const int*)d_in[1];
    const int*   dst    = (const int*)d_in[2];
    const float* W_lin  = (const float*)d_in[3];
    const float* b_lin  = (const float*)d_in[4];
    const float* W_self = (const float*)d_in[5];
    const float* b_self = (const float*)d_in[6];
    const float* bias   = (const float*)d_in[7];
    float* out = (float*)d_out;

    float* h = (float*)d_ws;   // [N_NODES, 64] f32 = 25.6 MB scratch

    // 1) zero h (float4-wide)
    {
        long long n4 = (long long)N_NODES * FEAT / 4;
        int threads = 256;
        long long blocks = (n4 + threads - 1) / threads;
        gc_zero_kernel<<<(unsigned)blocks, threads, 0, stream>>>(h, n4);
    }
    // 2) scatter-add over edges
    {
        long long total = (long long)NEDGE * FEAT;
        int threads = 256;
        long long blocks = (total + threads - 1) / threads;
        gc_scatter_kernel<<<(unsigned)blocks, threads, 0, stream>>>(x, src, dst, h);
    }
    // 3) fused WMMA dual-GEMM + bias (6250 M-strips, 8 waves/block)
    {
        int strips = N_NODES / 16;                 // 6250
        int blocks = (strips + 7) / 8;             // 782
        gc_gemm_kernel<<<blocks, 256, 0, stream>>>(h, x, W_lin, W_self,
                                                   b_lin, b_self, bias, out);
    }
}